// NodeCountingAutoencoder_36859409334287
// MI455X (gfx1250) — compile-verified
//
#include <hip/hip_runtime.h>
#include <stdint.h>
#include <stddef.h>

// Problem sizes (reference: B=1024, IN=512, HID=256)
#define B_N   1024
#define IN1   512
#define HID1  256

typedef __attribute__((ext_vector_type(4))) unsigned int v4u_t;
typedef __attribute__((ext_vector_type(8))) int          v8i_t;
typedef __attribute__((ext_vector_type(4))) int          v4i_t;

#if defined(__has_builtin)
#if __has_builtin(__builtin_amdgcn_tensor_load_to_lds) && __has_builtin(__builtin_amdgcn_s_wait_tensorcnt)
#define HAVE_TDM 1
#endif
#endif
#ifndef HAVE_TDM
#define HAVE_TDM 0
#endif

// ---------------------------------------------------------------------------
// TDM: issue a 2D tile load Global -> LDS.
//   tile_d0 = contiguous dim length (elements, 4B each)
//   tile_d1 = number of rows
//   stride_d0 = row stride in the global tensor (elements)
//   pad128: insert 1 DWORD of LDS padding every 128 DWORDs (row stride 129)
// Descriptor layout per CDNA5 ISA.
// This toolchain exposes the 6-arg builtin:
//   (uint32x4 g0, int32x8 g1, int32x4 g2, int32x4 g3, int32x8 extra, i32 cpol)
// ---------------------------------------------------------------------------
__device__ __forceinline__ void tdm_load_2d(uint32_t lds_byte_off, const void* gptr,
                                            uint32_t tile_d0, uint32_t tile_d1,
                                            uint32_t stride_d0, bool pad128) {
#if HAVE_TDM
  uint64_t ga = (uint64_t)(uintptr_t)gptr;
  v4u_t g0;
  g0.x = 1u;                                              // count=1 valid descriptor
  g0.y = lds_byte_off;                                    // lds_addr
  g0.z = (uint32_t)ga;                                    // global_addr[31:0]
  g0.w = (uint32_t)((ga >> 32) & 0x01FFFFFFull) | (2u << 30); // addr[56:32] | type=2
  uint32_t f = (2u << 16);                                // data_size=2 (4 bytes)
  if (pad128) f |= (1u << 20) | (6u << 22);               // pad_enable, interval=128 DW, amount=1 DW
  v8i_t g1;
  g1[0] = (int)f;
  g1[1] = (int)((tile_d0 & 0xffffu) << 16);                                   // tensor_dim0 lo16
  g1[2] = (int)(((tile_d0 >> 16) & 0xffffu) | ((tile_d1 & 0xffffu) << 16));   // td0 hi | tensor_dim1 lo
  g1[3] = (int)(((tile_d1 >> 16) & 0xffffu) | ((tile_d0 & 0xffffu) << 16));   // td1 hi | tile_dim0
  g1[4] = (int)(tile_d1 & 0xffffu);                                           // tile_dim1 (tile_dim2=0)
  g1[5] = (int)stride_d0;                                                     // tensor_dim0_stride lo32
  g1[6] = 0;                                                                  // stride hi | dim1_stride lo
  g1[7] = 0;
  v4i_t z4 = {0, 0, 0, 0};
  v8i_t z8 = {0, 0, 0, 0, 0, 0, 0, 0};
  __builtin_amdgcn_tensor_load_to_lds(g0, g1, z4, z4, z8, 0);
#else
  (void)lds_byte_off; (void)gptr; (void)tile_d0; (void)tile_d1; (void)stride_d0; (void)pad128;
#endif
}

__device__ __forceinline__ void tdm_wait(int n) {
#if HAVE_TDM
  if (n == 0) __builtin_amdgcn_s_wait_tensorcnt(0);
  else        __builtin_amdgcn_s_wait_tensorcnt(1);
#else
  (void)n;
#endif
}

// ---------------------------------------------------------------------------
// Prep kernel: bit-pack masks, compute Gumbel hard selections.
//   pk1[h][w]  (256 x 16 words)  from mask1[256][512]
//   pk2[j][w]  (512 x 8  words)  from mask2[512][256]
//   sel[row] = argmax(logits + -log(-log(clip(u))))  in {0,1}
// ---------------------------------------------------------------------------
__device__ __forceinline__ float gumbelf(float u) {
  u = fminf(fmaxf(u, 1e-10f), 1.0f);
  return -logf(-logf(u));
}

__global__ void prep_kernel(const int* __restrict__ mask1, const int* __restrict__ mask2,
                            const float* __restrict__ logits1, const float* __restrict__ u1,
                            const float* __restrict__ logits2, const float* __restrict__ u2,
                            uint32_t* __restrict__ pk1, uint32_t* __restrict__ pk2,
                            uint32_t* __restrict__ sel1, uint32_t* __restrict__ sel2) {
  int tid = blockIdx.x * blockDim.x + threadIdx.x;
  if (tid < 4096) {                     // mask1: 256 rows * 16 words; word tid covers mask1[tid*32 ..]
    const int* src = mask1 + (size_t)tid * 32;
    uint32_t m = 0;
    #pragma unroll
    for (int t = 0; t < 32; ++t) m |= (src[t] != 0 ? 1u : 0u) << t;
    pk1[tid] = m;
  } else if (tid < 8192) {              // mask2: 512 rows * 8 words
    int j = tid - 4096;
    const int* src = mask2 + (size_t)j * 32;
    uint32_t m = 0;
    #pragma unroll
    for (int t = 0; t < 32; ++t) m |= (src[t] != 0 ? 1u : 0u) << t;
    pk2[j] = m;
  } else if (tid < 8192 + HID1) {
    int r = tid - 8192;
    float s0 = logits1[r * 2 + 0] + gumbelf(u1[r * 2 + 0]);
    float s1 = logits1[r * 2 + 1] + gumbelf(u1[r * 2 + 1]);
    sel1[r] = (s1 > s0) ? 1u : 0u;      // argmax tie -> index 0
  } else if (tid < 8192 + HID1 + IN1) {
    int r = tid - 8192 - HID1;
    float s0 = logits2[r * 2 + 0] + gumbelf(u2[r * 2 + 0]);
    float s1 = logits2[r * 2 + 1] + gumbelf(u2[r * 2 + 1]);
    sel2[r] = (s1 > s0) ? 1u : 0u;
  }
}

// ---------------------------------------------------------------------------
// Masked min/max layer. Block = 256 threads (8 wave32), lanes = 32 batches.
// Each wave produces 4 output rows; grid = (B/32, OUT/32).
// Input is staged through LDS with double-buffered TDM loads of
// 32 batches x 128 elements per phase.
//   BMAJOR=true : src is [B][IN] row-major (x). LDS tile [b][i], row stride 129
//                 via TDM padding (conflict-free: bank = (129*lane+i)%64).
//                 Result stored transposed: dst[row][b] (coalesced).
//   BMAJOR=false: src is [IN][B] (h_t). LDS tile [i][b], naturally
//                 conflict-free. Result stored dst[b][row] (final output).
// ---------------------------------------------------------------------------
template<int IN_DIM, int OUT_DIM, bool BMAJOR>
__global__ __launch_bounds__(256)
void masked_minmax_layer(const float* __restrict__ src,
                         const uint32_t* __restrict__ pk,
                         const uint32_t* __restrict__ sel,
                         float* __restrict__ dst) {
  constexpr int PH     = IN_DIM / 128;               // phases
  constexpr int BUF_DW = BMAJOR ? (32 * 129) : (128 * 32);
  __shared__ float tile[2][BUF_DW + 4];

  const int lane = threadIdx.x & 31;
  const int wave = threadIdx.x >> 5;
  const int b0   = blockIdx.x * 32;
  const int b    = b0 + lane;
  const bool leader = (threadIdx.x == 0);
  const int row_base = blockIdx.y * 32 + wave * 4;

  auto issue = [&](int p, int buf) {
    uint32_t lds_off = (uint32_t)(uintptr_t)(const void*)(&tile[buf][0]);
    if (BMAJOR) {
      tdm_load_2d(lds_off, src + (size_t)b0 * IN_DIM + (size_t)p * 128,
                  /*tile_d0=*/128, /*tile_d1=*/32, /*stride=*/IN_DIM, /*pad128=*/true);
    } else {
      tdm_load_2d(lds_off, src + (size_t)p * 128 * B_N + b0,
                  /*tile_d0=*/32, /*tile_d1=*/128, /*stride=*/B_N, /*pad128=*/false);
    }
  };

#if !HAVE_TDM
  // Fallback staging (regular loads) so the file still compiles without TDM.
  auto stage = [&](int p, int buf) {
    for (int idx = threadIdx.x; idx < 32 * 128; idx += 256) {
      int row = idx >> 7, col = idx & 127;
      float v;
      if (BMAJOR) v = src[(size_t)(b0 + row) * IN_DIM + p * 128 + col];
      else        v = src[(size_t)(p * 128 + col) * B_N + b0 + row];
      if (BMAJOR) tile[buf][row * 129 + col] = v;
      else        tile[buf][col * 32 + row]  = v;
    }
  };
#endif

  if (leader) { issue(0, 0); if (PH > 1) issue(1, 1); }
#if !HAVE_TDM
  stage(0, 0); if (PH > 1) stage(1, 1);
#endif

  float amin[4], amax[4];
  #pragma unroll
  for (int r = 0; r < 4; ++r) { amin[r] = 1.0f; amax[r] = 0.0f; }

  const float nanv = __builtin_nanf("");

  #pragma unroll
  for (int p = 0; p < PH; ++p) {
    if (leader) tdm_wait(p == PH - 1 ? 0 : 1);   // in-order TDM completion
    __syncthreads();                             // tile[p&1] ready for all waves
    const float* tb = &tile[p & 1][0];

    #pragma unroll
    for (int w = 0; w < 4; ++w) {                // 4 mask words per 128-elt phase
      uint32_t m[4];
      #pragma unroll
      for (int r = 0; r < 4; ++r)
        m[r] = (uint32_t)__builtin_amdgcn_readfirstlane(
                  (int)pk[(size_t)(row_base + r) * (IN_DIM / 32) + p * 4 + w]);
      #pragma unroll
      for (int t = 0; t < 32; t += 2) {
        const int i0 = w * 32 + t;
        // one LDS read pair shared across 4 output rows
        float x0, x1;
        if (BMAJOR) { x0 = tb[lane * 129 + i0];  x1 = tb[lane * 129 + i0 + 1]; }
        else        { x0 = tb[i0 * 32 + lane];   x1 = tb[(i0 + 1) * 32 + lane]; }
        #pragma unroll
        for (int r = 0; r < 4; ++r) {
          // NaN trick: min_num/max_num ignore NaN -> one cndmask feeds both accs
          float a = ((m[r] >> t) & 1u) ? x0 : nanv;
          float c = ((m[r] >> (t + 1)) & 1u) ? x1 : nanv;
          amin[r] = fminf(fminf(amin[r], a), c);   // -> v_min3_num_f32
          amax[r] = fmaxf(fmaxf(amax[r], a), c);   // -> v_max3_num_f32
        }
      }
    }

    __syncthreads();                             // all waves done reading tile[p&1]
    if (leader && p + 2 < PH) issue(p + 2, p & 1);
#if !HAVE_TDM
    if (p + 2 < PH) { stage(p + 2, p & 1); }
#endif
  }

  #pragma unroll
  for (int r = 0; r < 4; ++r) {
    const int orow = row_base + r;
    const uint32_t s = sel[orow];                // 1 -> max node, 0 -> min node
    const float v = s ? amax[r] : amin[r];
    if (BMAJOR) dst[(size_t)orow * B_N + b] = v; // h stored transposed [HID][B]
    else        dst[(size_t)b * OUT_DIM + orow] = v;
  }
}

// ---------------------------------------------------------------------------
extern "C" void kernel_launch(void* const* d_in, const int* in_sizes, int n_in,
                              void* d_out, int out_size, void* d_ws, size_t ws_size,
                              hipStream_t stream) {
  (void)in_sizes; (void)n_in; (void)out_size; (void)ws_size;
  // setup_inputs order: x, logits1, u1, logits2, u2, mask1, mask2
  const float* x       = (const float*)d_in[0];
  const float* logits1 = (const float*)d_in[1];
  const float* u1      = (const float*)d_in[2];
  const float* logits2 = (const float*)d_in[3];
  const float* u2      = (const float*)d_in[4];
  const int*   mask1   = (const int*)d_in[5];
  const int*   mask2   = (const int*)d_in[6];

  // Workspace layout
  float*    h_t  = (float*)d_ws;                                   // [HID][B] = 1 MB
  uint32_t* pk1  = (uint32_t*)((char*)d_ws + (size_t)HID1 * B_N * sizeof(float));
  uint32_t* pk2  = pk1 + HID1 * (IN1 / 32);                        // 4096 words
  uint32_t* sel1 = pk2 + IN1 * (HID1 / 32);                        // 4096 words
  uint32_t* sel2 = sel1 + HID1;

  // 1) pack masks + compute hard selections (8960 threads)
  prep_kernel<<<35, 256, 0, stream>>>(mask1, mask2, logits1, u1, logits2, u2,
                                      pk1, pk2, sel1, sel2);

  // 2) layer 1: x[1024][512] -> h_t[256][1024]
  masked_minmax_layer<IN1, HID1, true><<<dim3(B_N / 32, HID1 / 32), 256, 0, stream>>>(
      x, pk1, sel1, h_t);

  // 3) layer 2: h_t[256][1024] -> out[1024][512]
  masked_minmax_layer<HID1, IN1, false><<<dim3(B_N / 32, IN1 / 32), 256, 0, stream>>>(
      h_t, pk2, sel2, (float*)d_out);
}